// LightGCN_12575664243390
// MI455X (gfx1250) — compile-verified
//
#include <hip/hip_runtime.h>

typedef __attribute__((ext_vector_type(2))) float v2f;
typedef __attribute__((ext_vector_type(8))) float v8f;

#define EMBED_DIM 64
#define TPB 256

// ---------------------------------------------------------------- utilities
__global__ void zero_kernel(float* __restrict__ p, int n) {
    int i = blockIdx.x * blockDim.x + threadIdx.x;
    if (i < n) p[i] = 0.0f;
}

// degree at target nodes (segment_sum of ones over dst)
__global__ void deg_kernel(const int* __restrict__ dst, float* __restrict__ deg, int E) {
    int e = blockIdx.x * blockDim.x + threadIdx.x;
    if (e < E) atomicAdd(&deg[dst[e]], 1.0f);
}

// deg -> deg^-1/2 in place (0 where deg==0)
__global__ void dinv_kernel(float* __restrict__ deg, int N) {
    int i = blockIdx.x * blockDim.x + threadIdx.x;
    if (i < N) {
        float d = deg[i];
        deg[i] = (d > 0.0f) ? rsqrtf(d) : 0.0f;
    }
}

// per-edge norm = dinv[src]*dinv[dst]
__global__ void norm_kernel(const int* __restrict__ src, const int* __restrict__ dst,
                            const float* __restrict__ dinv, float* __restrict__ normw, int E) {
    int e = blockIdx.x * blockDim.x + threadIdx.x;
    if (e < E) normw[e] = dinv[src[e]] * dinv[dst[e]];
}

// x = emb ; oemb = alpha[0]*emb ; xn = 0     (float4 vectorized over N*64/4)
__global__ void init_kernel(const float* __restrict__ emb, const float* __restrict__ alpha,
                            float* __restrict__ x, float* __restrict__ xn,
                            float* __restrict__ oemb, int n4) {
    int i = blockIdx.x * blockDim.x + threadIdx.x;
    if (i < n4) {
        float4 v = ((const float4*)emb)[i];
        float a0 = alpha[0];
        ((float4*)x)[i]    = v;
        ((float4*)oemb)[i] = make_float4(v.x * a0, v.y * a0, v.z * a0, v.w * a0);
        ((float4*)xn)[i]   = make_float4(0.f, 0.f, 0.f, 0.f);
    }
}

// one wave per edge: 32 lanes x float2 = 64-wide row; scatter-add into xn[dst]
__global__ void scatter_kernel(const float* __restrict__ x,
                               const int* __restrict__ src, const int* __restrict__ dst,
                               const float* __restrict__ normw,
                               float* __restrict__ xn, int E) {
    int t = blockIdx.x * blockDim.x + threadIdx.x;
    int e = t >> 5;
    if (e >= E) return;
    int lane = t & 31;
    float w = normw[e];
    const float* row = x + (size_t)src[e] * EMBED_DIM + lane * 2;
    float* orow      = xn + (size_t)dst[e] * EMBED_DIM + lane * 2;
    float v0 = row[0] * w;
    float v1 = row[1] * w;
    atomicAdd(orow + 0, v0);
    atomicAdd(orow + 1, v1);
}

// oemb += alpha[ai]*xn ; xz = 0  (prepares ping-pong buffer for next layer)
__global__ void accum_zero_kernel(float* __restrict__ oemb, const float* __restrict__ xn,
                                  const float* __restrict__ alpha, int ai,
                                  float* __restrict__ xz, int n4) {
    int i = blockIdx.x * blockDim.x + threadIdx.x;
    if (i < n4) {
        float a = alpha[ai];
        float4 o = ((float4*)oemb)[i];
        float4 v = ((const float4*)xn)[i];
        o.x += a * v.x; o.y += a * v.y; o.z += a * v.z; o.w += a * v.w;
        ((float4*)oemb)[i] = o;
        ((float4*)xz)[i] = make_float4(0.f, 0.f, 0.f, 0.f);
    }
}

// ---------------------------------------------------------------- WMMA dot
// One wave handles 16 edges. A = oemb[src rows] (16 x 64), B = oemb[dst rows]
// as columns (64 x 16). Accumulate 16 chunks of V_WMMA_F32_16X16X4_F32; the
// diagonal of the 16x16 f32 accumulator = the 16 edge dot products (full f32).
//
// 32-bit 16x4 A layout (ISA 7.12.2): lanes 0-15 hold M=0..15 with K=0,1 in
// v[0],v[1]; lanes 16-31 hold the same M with K=2,3. B (4x16) mirrored:
// lanes 0-15 hold N=0..15 with K=0,1; lanes 16-31 with K=2,3.
__global__ __launch_bounds__(TPB)
void dot_wmma_kernel(const float* __restrict__ oemb,
                     const int* __restrict__ src, const int* __restrict__ dst,
                     float* __restrict__ out, int num_groups) {
    int wave = (blockIdx.x * blockDim.x + threadIdx.x) >> 5;
    if (wave >= num_groups) return;              // wave-uniform: EXEC stays all-1s
    int lane  = threadIdx.x & 31;
    int ebase = wave * 16;
    int erow  = ebase + (lane & 15);
    int hi    = lane >> 4;                       // 0: K={0,1}, 1: K={2,3}

    const float* arow = oemb + (size_t)src[erow] * EMBED_DIM + 2 * hi;
    const float* brow = oemb + (size_t)dst[erow] * EMBED_DIM + 2 * hi;

    v8f acc = {};
#pragma unroll
    for (int k = 0; k < EMBED_DIM / 4; ++k) {
        v2f a = *(const v2f*)(arow + 4 * k);
        v2f b = *(const v2f*)(brow + 4 * k);
        // D = A(16x4) * B(4x16) + C, f32 in / f32 accum
        acc = __builtin_amdgcn_wmma_f32_16x16x4_f32(
            /*neg_a=*/false, a, /*neg_b=*/false, b,
            /*c_mod=*/(short)0, acc, /*reuse_a=*/false, /*reuse_b=*/false);
    }

    // Diagonal extraction (C/D 16x16 f32 layout):
    //  lanes 0-7  own D[m][m], m=lane,     in acc[lane]
    //  lanes 24-31 own D[m][m], m=lane-16, in acc[lane-24]
    if (lane < 8) {
        out[ebase + lane] = acc[lane];
    } else if (lane >= 24) {
        out[ebase + (lane - 16)] = acc[lane - 24];
    }
}

// scalar fallback for E % 16 tail (not hit for E = 1,000,000, kept for safety)
__global__ void dot_tail_kernel(const float* __restrict__ oemb,
                                const int* __restrict__ src, const int* __restrict__ dst,
                                float* __restrict__ out, int ebase, int E) {
    int e = ebase + blockIdx.x * blockDim.x + threadIdx.x;
    if (e < E) {
        const float* a = oemb + (size_t)src[e] * EMBED_DIM;
        const float* b = oemb + (size_t)dst[e] * EMBED_DIM;
        float s = 0.0f;
#pragma unroll
        for (int k = 0; k < EMBED_DIM; ++k) s += a[k] * b[k];
        out[e] = s;
    }
}

static inline size_t align256(size_t x) { return (x + 255) & ~(size_t)255; }

extern "C" void kernel_launch(void* const* d_in, const int* in_sizes, int n_in,
                              void* d_out, int out_size, void* d_ws, size_t ws_size,
                              hipStream_t stream) {
    const int*   edge_index = (const int*)d_in[0];   // [2, E] flat
    const float* emb        = (const float*)d_in[1]; // [N, 64]
    const float* alpha      = (const float*)d_in[2]; // [4]
    float* outv = (float*)d_out;

    const int E = in_sizes[0] / 2;
    const int N = in_sizes[1] / EMBED_DIM;
    const int* src = edge_index;
    const int* dst = edge_index + E;

    // -------- workspace carve-out (deg | norm | x_a | x_b | oemb) ~121 MB
    char* ws = (char*)d_ws;
    size_t off = 0;
    float* deg   = (float*)(ws + off); off += align256((size_t)N * 4);
    float* normw = (float*)(ws + off); off += align256((size_t)E * 4);
    float* x_a   = (float*)(ws + off); off += align256((size_t)N * EMBED_DIM * 4);
    float* x_b   = (float*)(ws + off); off += align256((size_t)N * EMBED_DIM * 4);
    float* oemb  = (float*)(ws + off); off += align256((size_t)N * EMBED_DIM * 4);
    (void)ws_size;

    const int n4 = N * EMBED_DIM / 4;

    // 1) gcn_norm
    zero_kernel<<<(N + TPB - 1) / TPB, TPB, 0, stream>>>(deg, N);
    deg_kernel<<<(E + TPB - 1) / TPB, TPB, 0, stream>>>(dst, deg, E);
    dinv_kernel<<<(N + TPB - 1) / TPB, TPB, 0, stream>>>(deg, N);
    norm_kernel<<<(E + TPB - 1) / TPB, TPB, 0, stream>>>(src, dst, deg, normw, E);

    // 2) out = alpha0*emb ; x_a = emb ; x_b = 0
    init_kernel<<<(n4 + TPB - 1) / TPB, TPB, 0, stream>>>(emb, alpha, x_a, x_b, oemb, n4);

    // 3) 3 propagation layers (ping-pong x_a <-> x_b)
    float* xc = x_a;
    float* xn = x_b;
    const int scat_threads = E * 32;
    for (int layer = 0; layer < 3; ++layer) {
        scatter_kernel<<<(scat_threads + TPB - 1) / TPB, TPB, 0, stream>>>(
            xc, src, dst, normw, xn, E);
        // out += alpha[layer+1]*xn ; zero old xc so it can serve as next xn
        accum_zero_kernel<<<(n4 + TPB - 1) / TPB, TPB, 0, stream>>>(
            oemb, xn, alpha, layer + 1, xc, n4);
        float* t = xc; xc = xn; xn = t;
    }

    // 4) edge-wise dot products via V_WMMA_F32_16X16X4_F32 (16 edges / wave)
    const int num_groups = E / 16;
    if (num_groups > 0) {
        int waves_per_block = TPB / 32;
        int blocks = (num_groups + waves_per_block - 1) / waves_per_block;
        dot_wmma_kernel<<<blocks, TPB, 0, stream>>>(oemb, src, dst, outv, num_groups);
    }
    const int rem = E % 16;
    if (rem > 0) {
        dot_tail_kernel<<<(rem + TPB - 1) / TPB, TPB, 0, stream>>>(
            oemb, src, dst, outv, num_groups * 16, E);
    }
}